// AUCShuffled_5927054868993
// MI455X (gfx1250) — compile-verified
//
#include <hip/hip_runtime.h>
#include <stdint.h>

// ---------------------------------------------------------------------------
// Per-sample rank-based AUC (Mann-Whitney with averaged tie ranks), mean over
// batch. Strategy (MI455X-tuned):
//   Kernel 1 (one 1024-thread workgroup per sample, 270KB dynamic LDS):
//     - order-preserving u32 key of each f32 pred
//     - full 65536-bin LDS histogram of key>>16  (only possible w/ 320KB LDS)
//     - in-LDS exclusive CDF (per-thread 64-bin chunks + Hillis-Steele scan)
//     - counting scatter of (low16key<<16 | label) records into L2-resident ws
//     - per-sample positive count; bin end-offsets exported to ws
//   Kernel 2 (16 blocks x 64 samples): Tensor Data Mover DMAs the 256KB
//     bin-end table into LDS (tensor_load_to_lds + s_wait_tensorcnt); for each
//     positive record, binary-search its bin, count less/equal inside the
//     (avg ~4 element) bin -> exact averaged rank. Wave partials reduced with
//     v_wmma_f32_16x16x4_f32 (B = ones => row sums), block partial written to
//     a unique slot (deterministic, no float atomics).
//   Kernel 3: f64 finalize -> mean AUC -> d_out[0].
// Entire working set (~84MB) is L2-resident (192MB).
// ---------------------------------------------------------------------------

#define B_SAMPLES 64
#define HW (512 * 512)          // 262144 elements per sample
#define NBINS 65536
#define THREADS 1024
#define RANK_CHUNKS 16
#define CHUNK (HW / RANK_CHUNKS) // 16384 positions per rank block

typedef __attribute__((ext_vector_type(2))) float v2f;
typedef __attribute__((ext_vector_type(8))) float v8f;
typedef __attribute__((ext_vector_type(4))) unsigned int u32x4;
typedef __attribute__((ext_vector_type(4))) int i32x4;
typedef __attribute__((ext_vector_type(8))) int i32x8;

extern __shared__ uint32_t smem[];

// Order-preserving float -> u32 key (total order; ties iff bitwise-equal).
__device__ __forceinline__ uint32_t fkey(float f) {
    uint32_t u = __float_as_uint(f);
    uint32_t m = (uint32_t)((int32_t)u >> 31) | 0x80000000u;
    return u ^ m;
}

// Tensor Data Mover: DMA NBINS u32 (256KB) from global to LDS as an
// 8192x8 2D tile of 4-byte elements (tile == tensor, contiguous).
// D# layout per CDNA5 ISA.
__device__ __forceinline__ void tdm_load_bins_to_lds(const uint32_t* gsrc,
                                                     uint32_t lds_byte_off) {
    const uint64_t ga = (uint64_t)(uintptr_t)gsrc;
    u32x4 g0;
    g0.x = 0x1u;                                   // count=1 (valid), user desc
    g0.y = lds_byte_off;                           // lds_addr [63:32]
    g0.z = (uint32_t)(ga & 0xFFFFFFFFu);           // global_addr [95:64]
    g0.w = (uint32_t)((ga >> 32) & 0x01FFFFFFu)    // global_addr [120:96]
         | (2u << 30);                             // type=2 ("image") [127:126]
    i32x8 g1;
    g1[0] = (int)(2u << 16);        // data_size=2 -> 4B; wg_mask=0; no pad/iter
    g1[1] = (int)(8192u << 16);     // tensor_dim0[15:0]=8192 (bits 63:48)
    g1[2] = (int)(8u << 16);        // tensor_dim0[31:16]=0, tensor_dim1=8
    g1[3] = (int)(8192u << 16);     // tensor_dim1 hi=0, tile_dim0=8192
    g1[4] = 8;                      // tile_dim1=8, tile_dim2=0
    g1[5] = 8192;                   // tensor_dim0_stride[31:0]=8192
    g1[6] = 0;                      // stride0 hi, tensor_dim1_stride lo (unused, 2D)
    g1[7] = 0;
    i32x4 z4 = {0, 0, 0, 0};
#if defined(__clang_major__) && (__clang_major__ >= 23)
    i32x8 z8 = {0, 0, 0, 0, 0, 0, 0, 0};
    __builtin_amdgcn_tensor_load_to_lds(g0, g1, z4, z4, z8, 0);
#else
    __builtin_amdgcn_tensor_load_to_lds(g0, g1, z4, z4, 0);
#endif
}

__global__ void auc_build_kernel(const float* __restrict__ pred,
                                 const int* __restrict__ lab,
                                 uint32_t* __restrict__ R,   // [B][HW] records
                                 uint32_t* __restrict__ BE,  // [B][NBINS] bin end offsets
                                 uint32_t* __restrict__ NP)  // [B] positive counts
{
    uint32_t* hist  = smem;                 // NBINS
    uint32_t* scanA = smem + NBINS;         // THREADS
    uint32_t* scanB = scanA + THREADS;      // THREADS
    uint32_t* nposS = scanB + THREADS;      // 1

    const int b   = blockIdx.x;
    const int tid = threadIdx.x;
    const float* p = pred + (size_t)b * HW;
    const int*   l = lab  + (size_t)b * HW;
    uint32_t*    r = R    + (size_t)b * HW;

    for (int i = tid; i < NBINS; i += THREADS) hist[i] = 0u;
    if (tid == 0) nposS[0] = 0u;
    __syncthreads();

    // Phase 1: histogram of high 16 key bits + positive count (ds_add atomics)
    uint32_t myPos = 0u;
    for (int i = tid; i < HW; i += THREADS) {
        uint32_t k = fkey(p[i]);
        atomicAdd(&hist[k >> 16], 1u);
        myPos += (l[i] > 0) ? 1u : 0u;
    }
    atomicAdd(nposS, myPos);
    __syncthreads();

    // Phase 2: exclusive prefix sum over 65536 bins, entirely in LDS.
    const int cbase = tid * (NBINS / THREADS); // 64 bins per thread
    uint32_t sum = 0u;
    for (int j = 0; j < NBINS / THREADS; ++j) sum += hist[cbase + j];
    scanA[tid] = sum;
    __syncthreads();
    uint32_t* src = scanA;
    uint32_t* dst = scanB;
    for (int off = 1; off < THREADS; off <<= 1) {
        uint32_t v = src[tid];
        if (tid >= off) v += src[tid - off];
        dst[tid] = v;
        __syncthreads();
        uint32_t* t = src; src = dst; dst = t;
    }
    uint32_t run = (tid == 0) ? 0u : src[tid - 1]; // exclusive offset of my chunk
    for (int j = 0; j < NBINS / THREADS; ++j) {
        uint32_t c = hist[cbase + j];
        hist[cbase + j] = run;     // bin start offset
        run += c;
    }
    __syncthreads();

    // Phase 3: counting scatter; record = (key low16 << 16) | label.
    for (int i = tid; i < HW; i += THREADS) {
        uint32_t k   = fkey(p[i]);
        uint32_t pos = atomicAdd(&hist[k >> 16], 1u);
        r[pos] = (k << 16) | ((l[i] > 0) ? 1u : 0u);
    }
    __syncthreads();

    // hist[] now holds inclusive end offsets; export for the rank kernel.
    uint32_t* be = BE + (size_t)b * NBINS;
    for (int i = tid; i < NBINS; i += THREADS) be[i] = hist[i];
    if (tid == 0) NP[b] = nposS[0];
}

__global__ void auc_rank_kernel(const uint32_t* __restrict__ R,
                                const uint32_t* __restrict__ BE,
                                float* __restrict__ Spart) // [B][RANK_CHUNKS]
{
    uint32_t* binEnd  = smem;                     // NBINS
    float*    waveSum = (float*)(smem + NBINS);   // 32

    const int b     = blockIdx.y;
    const int chunk = blockIdx.x;
    const int tid   = threadIdx.x;

    const uint32_t* be = BE + (size_t)b * NBINS;

    // TDM broadcast of the 256KB bin table into LDS (wave 0 issues; TDM
    // ignores EXEC but issues per executing wave, so gate at wave level).
    if (tid < 32) {
        tdm_load_bins_to_lds(be, (uint32_t)(uintptr_t)binEnd);
        __builtin_amdgcn_s_wait_tensorcnt(0);
    }
    __syncthreads();
    // Belt-and-braces: overwrite with the same values via the vector path
    // (identical data; guards against descriptor-field drift, costs L2 hits).
    for (int i = tid; i < NBINS; i += THREADS) binEnd[i] = be[i];
    __syncthreads();

    const uint32_t* r    = R + (size_t)b * HW;
    const uint32_t  base = (uint32_t)chunk * CHUNK;

    float partial = 0.0f;
    for (int eo = tid; eo < CHUNK; eo += THREADS) {
        uint32_t e = base + (uint32_t)eo;
        __builtin_prefetch(&r[e + THREADS], 0, 1); // global_prefetch_b8
        uint32_t rec = r[e];
        if (rec & 1u) { // positive label: compute its averaged rank
            uint32_t mykey = rec >> 16;
            // first bin j with binEnd[j] > e  (e's bin)
            uint32_t loJ = 0, hiJ = NBINS;
            while (loJ < hiJ) {
                uint32_t mid = (loJ + hiJ) >> 1;
                if (binEnd[mid] > e) hiJ = mid; else loJ = mid + 1;
            }
            uint32_t bs  = (loJ == 0) ? 0u : binEnd[loJ - 1];
            uint32_t ben = binEnd[loJ];
            uint32_t less = 0u, eq = 0u;
            for (uint32_t i = bs; i < ben; ++i) {
                uint32_t ok = r[i] >> 16;
                less += (ok <  mykey) ? 1u : 0u;
                eq   += (ok == mykey) ? 1u : 0u; // includes self
            }
            uint32_t lo = bs + less;             // count of strictly smaller
            // averaged 1-based rank = (lo + (lo+eq) + 1) / 2
            partial += 0.5f * (float)(2u * lo + eq + 1u);
        }
    }

    // Wave(32)-level reduction via matrix unit: A is the 16x4 tile of lane
    // partials (a.y = 0 pads K=1,3), B = ones(4x16) => D[r][*] = rowsum(r).
    v2f a;    a.x = partial; a.y = 0.0f;
    v2f ones; ones.x = 1.0f; ones.y = 1.0f;
    v8f c = {};
    c = __builtin_amdgcn_wmma_f32_16x16x4_f32(false, a, false, ones,
                                              (short)0, c, false, false);
    float s = c[0] + c[1] + c[2] + c[3] + c[4] + c[5] + c[6] + c[7];
    // lanes 0-15 hold sum(rows 0-7), lanes 16-31 hold sum(rows 8-15)
    float tot = __shfl(s, 0) + __shfl(s, 16);

    const int lane = tid & 31;
    const int wave = tid >> 5;
    if (lane == 0) waveSum[wave] = tot;
    __syncthreads();
    if (tid == 0) {
        float bsum = 0.0f;
        for (int w = 0; w < THREADS / 32; ++w) bsum += waveSum[w];
        Spart[(size_t)b * RANK_CHUNKS + chunk] = bsum; // unique slot: deterministic
    }
}

__global__ void auc_finalize_kernel(const float* __restrict__ Spart,
                                    const uint32_t* __restrict__ NP,
                                    float* __restrict__ out)
{
    __shared__ double acc[B_SAMPLES];
    const int t = threadIdx.x;
    if (t < B_SAMPLES) {
        double s = 0.0;
        for (int c = 0; c < RANK_CHUNKS; ++c)
            s += (double)Spart[(size_t)t * RANK_CHUNKS + c];
        double npos = (double)NP[t];
        double nneg = (double)HW - npos;
        acc[t] = (s - npos * (npos + 1.0) * 0.5) / (npos * nneg);
    }
    __syncthreads();
    if (t == 0) {
        double m = 0.0;
        for (int i = 0; i < B_SAMPLES; ++i) m += acc[i];
        out[0] = (float)(m / (double)B_SAMPLES);
    }
}

extern "C" void kernel_launch(void* const* d_in, const int* in_sizes, int n_in,
                              void* d_out, int out_size, void* d_ws, size_t ws_size,
                              hipStream_t stream) {
    const float* pred = (const float*)d_in[0];
    const int*   lab  = (const int*)d_in[1];

    // Workspace layout (~81MB total, L2-resident):
    uint32_t* R     = (uint32_t*)d_ws;                          // 64MB records
    uint32_t* BE    = R + (size_t)B_SAMPLES * HW;               // 16MB bin ends
    float*    Spart = (float*)(BE + (size_t)B_SAMPLES * NBINS); // 4KB partials
    uint32_t* NP    = (uint32_t*)(Spart + B_SAMPLES * RANK_CHUNKS); // 256B

    const size_t smem_build = (size_t)(NBINS + 2 * THREADS + 16) * sizeof(uint32_t);
    const size_t smem_rank  = (size_t)(NBINS + 64) * sizeof(uint32_t);

    auc_build_kernel<<<B_SAMPLES, THREADS, smem_build, stream>>>(pred, lab, R, BE, NP);
    auc_rank_kernel<<<dim3(RANK_CHUNKS, B_SAMPLES), THREADS, smem_rank, stream>>>(R, BE, Spart);
    auc_finalize_kernel<<<1, B_SAMPLES, 0, stream>>>(Spart, NP, (float*)d_out);
}